// DetectionLoss_41171556499948
// MI455X (gfx1250) — compile-verified
//
#include <hip/hip_runtime.h>
#include <hip/hip_bf16.h>
#include <math.h>

#define NUM_B   8
#define N_ANCH  19200     // 3*80*80
#define M_TGT   100
#define C_CLS   80

typedef __attribute__((ext_vector_type(2))) float v2f;
typedef __attribute__((ext_vector_type(8))) float v8f;

#if defined(__gfx1250__)

typedef __attribute__((ext_vector_type(4))) unsigned int u32x4;
typedef __attribute__((ext_vector_type(8))) int          i32x8;
typedef __attribute__((ext_vector_type(4))) int          i32x4;

#if __has_builtin(__builtin_amdgcn_tensor_load_to_lds)
#define USE_TDM 1
// 1-D TDM copy of `ndw` dwords global -> LDS. D# bitfields per CDNA5 ISA ch.8:
// group0: [1:0]=count=1, [63:32]=lds_addr, [120:64]=global_addr, [127:126]=type=2
// group1: [17:16]=data_size(4B), [79:48]=tensor_dim0, [111:80]=tensor_dim1=1,
//         [127:112]=tile_dim0, [143:128]=tile_dim1=1, [207:160]=dim0_stride
__device__ __forceinline__ void tdm_load_1d(const void* gsrc, void* ldst, unsigned ndw) {
  unsigned long long ga = (unsigned long long)gsrc;
  unsigned la = (unsigned)(unsigned long long)ldst;   // low 32 bits = LDS byte offset
  u32x4 g0;
  g0[0] = 1u;                                             // count=1 (valid user D#)
  g0[1] = la;                                             // lds_addr
  g0[2] = (unsigned)ga;                                   // global_addr[31:0]
  g0[3] = (unsigned)((ga >> 32) & 0x01FFFFFFu) | (2u << 30); // addr[56:32] | type=2
  i32x8 g1;
  g1[0] = (int)(2u << 16);                                // data_size = 4 bytes
  g1[1] = (int)((ndw & 0xFFFFu) << 16);                   // tensor_dim0[15:0]
  g1[2] = (int)(((ndw >> 16) & 0xFFFFu) | (1u << 16));    // tensor_dim0[31:16], tensor_dim1=1
  g1[3] = (int)((ndw & 0xFFFFu) << 16);                   // tile_dim0 = ndw
  g1[4] = 1;                                              // tile_dim1 = 1, tile_dim2 = 0
  g1[5] = (int)ndw;                                       // tensor_dim0_stride[31:0]
  g1[6] = 0;
  g1[7] = 0;
  i32x4 z4 = {0, 0, 0, 0};                                // groups 2/3 unused (<=2D)
  i32x8 z8 = {0, 0, 0, 0, 0, 0, 0, 0};
  __builtin_amdgcn_tensor_load_to_lds(g0, g1, z4, z4, z8, 0);
}
#endif

// 32-lane sum via V_WMMA_F32_16X16X4_F32:
// A[16x4]: lanes 0-15 -> A[lane][0]=x, A[lane][1]=0; lanes 16-31 -> A[lane-16][2]=x, [3]=0.
// B = ones(4x16) => D[m][n] = x[m] + x[m+16] (replicated across columns).
// Lane L<16 holds D rows 0..7, L>=16 rows 8..15; sum 8 regs, then swapx16-add.
__device__ __forceinline__ float wave_sum(float x) {
  v2f a;  a[0] = x;   a[1] = 0.0f;
  v2f bm; bm[0] = 1.0f; bm[1] = 1.0f;
  v8f c = {0.f, 0.f, 0.f, 0.f, 0.f, 0.f, 0.f, 0.f};
  v8f d = __builtin_amdgcn_wmma_f32_16x16x4_f32(false, a, false, bm, (short)0, c, false, false);
  float s = ((d[0] + d[1]) + (d[2] + d[3])) + ((d[4] + d[5]) + (d[6] + d[7]));
  s += __shfl_xor(s, 16, 32);
  return s;   // total of all 32 lanes, in every lane
}

#else  // host pass / non-gfx1250: shuffle fallback (never codegen'd for device)
__device__ __forceinline__ float wave_sum(float x) {
  for (int off = 16; off > 0; off >>= 1) x += __shfl_xor(x, off, 32);
  return x;
}
#endif

__global__ void det_loss_zero(float* acc) { acc[threadIdx.x] = 0.0f; }

__global__ __launch_bounds__(256) void det_loss_main(
    const float* __restrict__ obj,      // [B, N]
    const float* __restrict__ boxes,    // [B, N, 4] (cx,cy,w,h)
    const float* __restrict__ cls,      // [B, N, 80]
    const float* __restrict__ tboxes,   // [B, 100, 4]
    const int*   __restrict__ tlabels,  // [B, 100]
    float* __restrict__ acc)            // [B, 4]: bce_sum, pos_cnt, bbox_sum, fl_sum
{
  __shared__ float stb[M_TGT * 4];
  __shared__ int   stl[M_TGT];
  const int b = blockIdx.y;
  const int i = blockIdx.x * 256 + threadIdx.x;   // grid tiles N exactly

  const float* gtb = tboxes  + (size_t)b * (M_TGT * 4);
  const int*   gtl = tlabels + (size_t)b * M_TGT;
#if defined(USE_TDM)
  if (threadIdx.x < 32) {                 // wave 0 issues DMA, waits, barrier releases rest
    tdm_load_1d(gtb, stb, M_TGT * 4);
    tdm_load_1d(gtl, stl, M_TGT);
    __builtin_amdgcn_s_wait_tensorcnt(0);
  }
#else
  for (int t = (int)threadIdx.x; t < M_TGT * 4; t += 256) stb[t] = gtb[t];
  for (int t = (int)threadIdx.x; t < M_TGT;     t += 256) stl[t] = gtl[t];
#endif
  __syncthreads();

  // ---- anchor box, IoU sweep over 100 targets (LDS broadcast reads) ----
  const float4 bx = ((const float4*)boxes)[(size_t)b * N_ANCH + i];
  const float w = bx.z, h = bx.w;
  const float l1x = bx.x - 0.5f * w, h1x = bx.x + 0.5f * w;
  const float l1y = bx.y - 0.5f * h, h1y = bx.y + 0.5f * h;
  const float area1 = w * h;

  float best = -1.0f; int bidx = 0;
#pragma unroll 5
  for (int m = 0; m < M_TGT; ++m) {
    const float tcx = stb[4 * m + 0], tcy = stb[4 * m + 1];
    const float tw  = stb[4 * m + 2], th  = stb[4 * m + 3];
    float iw = fminf(h1x, tcx + 0.5f * tw) - fmaxf(l1x, tcx - 0.5f * tw);
    float ih = fminf(h1y, tcy + 0.5f * th) - fmaxf(l1y, tcy - 0.5f * th);
    iw = fmaxf(iw, 0.0f); ih = fmaxf(ih, 0.0f);
    const float inter = iw * ih;
    const float uni   = area1 + tw * th - inter;
    const float iou   = inter / (uni + 1e-6f);
    if (iou > best) { best = iou; bidx = m; }   // first-max == jnp.argmax
  }
  const bool pos = (best >= 0.5f);
  const float posf = pos ? 1.0f : 0.0f;

  const float* crow = cls + ((size_t)b * N_ANCH + i) * C_CLS;
  if (pos) __builtin_prefetch(crow, 0, 1);      // global_prefetch_b8; cover latency

  // ---- objectness BCE (torch BCELoss clamp at -100) ----
  const float p = obj[(size_t)b * N_ANCH + i];
  const float bce = pos ? -fmaxf(logf(p), -100.0f) : -fmaxf(log1pf(-p), -100.0f);

  // ---- GIoU + focal loss: positives only (negatives are exactly zeroed in ref) ----
  float gterm = 0.0f, flsum = 0.0f;
  if (pos) {
    const float tcx = stb[4 * bidx + 0], tcy = stb[4 * bidx + 1];
    const float tw  = stb[4 * bidx + 2], th  = stb[4 * bidx + 3];
    const float l2x = tcx - 0.5f * tw, h2x = tcx + 0.5f * tw;
    const float l2y = tcy - 0.5f * th, h2y = tcy + 0.5f * th;
    const float iw = fmaxf(fminf(h1x, h2x) - fmaxf(l1x, l2x), 0.0f);
    const float ih = fmaxf(fminf(h1y, h2y) - fmaxf(l1y, l2y), 0.0f);
    const float inter = iw * ih;
    const float uni   = area1 + tw * th - inter;
    const float iou   = inter / (uni + 1e-6f);
    const float ew = fmaxf(fmaxf(h1x, h2x) - fminf(l1x, l2x), 0.0f);
    const float eh = fmaxf(fmaxf(h1y, h2y) - fminf(l1y, l2y), 0.0f);
    const float enc = ew * eh;
    gterm = 1.0f - (iou - (enc - uni) / (enc + 1e-6f));

    const int t = stl[bidx];
    const float4* c4 = (const float4*)crow;     // 320B row, 16B aligned
#pragma unroll 4
    for (int q = 0; q < C_CLS / 4; ++q) {
      const float4 v = c4[q];
      const float sv[4] = {v.x, v.y, v.z, v.w};
#pragma unroll
      for (int e = 0; e < 4; ++e) {
        const float s  = sv[e];
        const bool  oh = ((4 * q + e) == t);
        const float ce = fmaxf(s, 0.0f) - (oh ? s : 0.0f) + log1pf(expf(-fabsf(s)));
        const float pp = 1.0f / (1.0f + expf(-s));
        const float pt = oh ? pp : 1.0f - pp;
        const float at = oh ? 0.25f : 0.75f;
        const float om = 1.0f - pt;
        flsum += at * om * om * ce;
      }
    }
  }

  // ---- per-wave WMMA reduction (uniform control flow, EXEC all-ones), then atomics ----
  const float s_bce = wave_sum(bce);
  const float s_cnt = wave_sum(posf);
  const float s_bb  = wave_sum(gterm);
  const float s_fl  = wave_sum(flsum);
  if ((threadIdx.x & 31u) == 0u) {
    atomicAdd(&acc[b * 4 + 0], s_bce);
    atomicAdd(&acc[b * 4 + 1], s_cnt);
    atomicAdd(&acc[b * 4 + 2], s_bb);
    atomicAdd(&acc[b * 4 + 3], s_fl);
  }
}

__global__ void det_loss_finalize(const float* __restrict__ acc, float* __restrict__ out) {
  float obj_t = 0.0f, bbox_t = 0.0f, cls_t = 0.0f, npos = 0.0f;
  for (int b = 0; b < NUM_B; ++b) {
    const float bce = acc[4 * b + 0], cnt = acc[4 * b + 1];
    const float bb  = acc[4 * b + 2], fl  = acc[4 * b + 3];
    obj_t  += bce * (1.0f * cnt + 0.5f * ((float)N_ANCH - cnt));  // λobj, λnoobj
    bbox_t += bb;
    cls_t  += fl / fmaxf(cnt * (float)C_CLS, 1.0f);
    npos   += cnt;
  }
  npos = fmaxf(npos, 1.0f);
  out[0] = obj_t / (float)NUM_B + 5.0f * bbox_t / npos + cls_t / (float)NUM_B;
}

extern "C" void kernel_launch(void* const* d_in, const int* in_sizes, int n_in,
                              void* d_out, int out_size, void* d_ws, size_t ws_size,
                              hipStream_t stream) {
  const float* obj    = (const float*)d_in[0];  // [8,3,80,80,1]
  const float* boxes  = (const float*)d_in[1];  // [8,3,80,80,4]
  const float* cls    = (const float*)d_in[2];  // [8,3,80,80,80]
  const float* tb     = (const float*)d_in[3];  // [8,100,4]
  const int*   tl     = (const int*)d_in[4];    // [8,100]
  float* out = (float*)d_out;
  float* acc = (float*)d_ws;                    // 32 floats

  det_loss_zero<<<1, NUM_B * 4, 0, stream>>>(acc);
  dim3 grid(N_ANCH / 256, NUM_B);
  det_loss_main<<<grid, 256, 0, stream>>>(obj, boxes, cls, tb, tl, acc);
  det_loss_finalize<<<1, 1, 0, stream>>>(acc, out);
}